// LengthRegulator_80006650790022
// MI455X (gfx1250) — compile-verified
//
#include <hip/hip_runtime.h>
#include <hip/hip_bf16.h>
#include <stdint.h>

// LengthRegulator: pure-bandwidth gather. B=32, T=1024, D=384 (fp32).
// Roofline: ~50MB read (L2-resident) + ~180MB NT write -> ~10us @ 23.3 TB/s.
// No WMMA: zero FLOPs. CDNA5 paths used: async global->LDS staging of the
// cumsum row (+ s_wait_asynccnt), b128 vector memory ops, LDS binary search,
// non-temporal stores to keep encodings resident in the 192MB L2.

#define LR_B 32

typedef float f4 __attribute__((ext_vector_type(4)));
typedef int   v4i __attribute__((ext_vector_type(4)));

#if defined(__AMDGCN__) && __has_builtin(__builtin_amdgcn_global_load_async_to_lds_b128)
#define LR_HAVE_ASYNC_LDS 1
typedef __attribute__((address_space(1))) v4i* gv4i_p;  // global (AS1)
typedef __attribute__((address_space(3))) v4i* lv4i_p;  // LDS (AS3)
#else
#define LR_HAVE_ASYNC_LDS 0
#endif

// ---------------------------------------------------------------------------
// Kernel 1: per-batch inclusive cumsum of durations. One wave32 per batch.
// Each lane sequentially scans a 32-element chunk, then a shuffle-based
// wave scan of the chunk totals produces the exclusive offsets.
// ---------------------------------------------------------------------------
__global__ __launch_bounds__(32) void lr_cumsum_kernel(const int* __restrict__ dur,
                                                       int* __restrict__ csum,
                                                       int T) {
  const int b    = blockIdx.x;
  const int lane = threadIdx.x;            // 0..31
  const int ch   = T >> 5;                 // 32 elements per lane when T==1024
  const int base = b * T + lane * ch;

  int vals[32];
  int s = 0;
  if (ch == 32) {
#pragma unroll
    for (int i = 0; i < 32; ++i) {
      s += dur[base + i];
      vals[i] = s;
    }
  } else {
    for (int i = 0; i < ch && i < 32; ++i) {
      s += dur[base + i];
      vals[i] = s;
    }
  }

  // Inclusive wave32 scan of per-lane sums -> exclusive prefix for this lane.
  int x = s;
#pragma unroll
  for (int off = 1; off < 32; off <<= 1) {
    int y = __shfl_up(x, off, 32);
    if (lane >= off) x += y;
  }
  const int excl = x - s;

  if (ch == 32) {
#pragma unroll
    for (int i = 0; i < 32; ++i) csum[base + i] = vals[i] + excl;
  } else {
    for (int i = 0; i < ch && i < 32; ++i) csum[base + i] = vals[i] + excl;
  }
}

// ---------------------------------------------------------------------------
// Kernel 2: gather. grid=(B, SPLIT), block=256 (8 waves). The block stages its
// batch's csum row (4KB) in LDS (async global->LDS when available), then each
// wave owns output rows p in a strided loop: wave-uniform binary search in
// LDS, then a 1536B row copy as 3x b128 loads + 3x non-temporal b128 stores.
// Rows past totals[b] get NT zero stores (harness does not re-poison d_out).
// ---------------------------------------------------------------------------
__global__ __launch_bounds__(256) void lr_gather_kernel(const float* __restrict__ enc,
                                                        const int* __restrict__ csum,
                                                        float* __restrict__ out,
                                                        int T, int D, int max_len) {
  __shared__ __align__(16) int s_csum[1024];
  const int b   = blockIdx.x;
  const int tid = threadIdx.x;
  const int* crow = csum + (size_t)b * T;

#if LR_HAVE_ASYNC_LDS
  {
    const int i = tid * 4;                 // 256 threads * 16B == 4KB == T ints
    if (i < T) {
      // Generic-space reinterpret + const drop, then pure addrspacecasts.
      v4i* gsrc_gen = const_cast<v4i*>(reinterpret_cast<const v4i*>(crow + i));
      v4i* ldst_gen = reinterpret_cast<v4i*>(&s_csum[i]);
      __builtin_amdgcn_global_load_async_to_lds_b128(
          (gv4i_p)gsrc_gen, (lv4i_p)ldst_gen, 0, 0);
    }
#if __has_builtin(__builtin_amdgcn_s_wait_asynccnt)
    __builtin_amdgcn_s_wait_asynccnt(0);
#else
    asm volatile("s_wait_asynccnt 0" ::: "memory");
#endif
  }
#else
  for (int i = tid; i < T; i += 256) s_csum[i] = crow[i];
#endif
  __syncthreads();

  const int total = s_csum[T - 1];
  const int lane  = tid & 31;
  const int wave  = tid >> 5;
  const int vpr   = D >> 2;                          // float4 per row (96)
  const f4* encB  = (const f4*)enc + (size_t)b * T * vpr;
  f4* outB        = (f4*)out + (size_t)b * max_len * vpr;
  const int pstride = gridDim.y * 8;

  for (int p = blockIdx.y * 8 + wave; p < max_len; p += pstride) {
    f4* dst = outB + (size_t)p * vpr;

    if (p >= total) {
      const f4 z = (f4)0.0f;
      if (vpr == 96) {
        __builtin_nontemporal_store(z, dst + lane);
        __builtin_nontemporal_store(z, dst + lane + 32);
        __builtin_nontemporal_store(z, dst + lane + 64);
      } else {
        for (int v = lane; v < vpr; v += 32) __builtin_nontemporal_store(z, dst + v);
      }
      continue;
    }

    // searchsorted(csum, p, side='right'): first j with csum[j] > p.
    int lo = 0, hi = T;
    while (lo < hi) {
      int mid = (lo + hi) >> 1;
      if (s_csum[mid] <= p) lo = mid + 1; else hi = mid;
    }
    const int idx = (lo < T) ? lo : (T - 1);
    const f4* src = encB + (size_t)idx * vpr;

    if (vpr == 96) {
      f4 a0 = src[lane];
      f4 a1 = src[lane + 32];
      f4 a2 = src[lane + 64];
      __builtin_nontemporal_store(a0, dst + lane);
      __builtin_nontemporal_store(a1, dst + lane + 32);
      __builtin_nontemporal_store(a2, dst + lane + 64);
    } else {
      for (int v = lane; v < vpr; v += 32) {
        f4 a = src[v];
        __builtin_nontemporal_store(a, dst + v);
      }
    }
  }
}

// ---------------------------------------------------------------------------
extern "C" void kernel_launch(void* const* d_in, const int* in_sizes, int n_in,
                              void* d_out, int out_size, void* d_ws, size_t ws_size,
                              hipStream_t stream) {
  const float* enc = (const float*)d_in[0];   // (B, T, D) fp32
  const int*   dur = (const int*)d_in[1];     // (B, T) int32
  float*       out = (float*)d_out;           // (B, max_len, D) fp32

  const int BT      = in_sizes[1];
  const int B       = LR_B;
  const int T       = BT / B;                 // 1024
  const int D       = in_sizes[0] / BT;       // 384
  const int max_len = (B * D) ? out_size / (B * D) : 0;

  int* csum = (int*)d_ws;                     // B*T ints = 128KB scratch

  lr_cumsum_kernel<<<dim3(B), dim3(32), 0, stream>>>(dur, csum, T);
  lr_gather_kernel<<<dim3(B, 32), dim3(256), 0, stream>>>(enc, csum, out, T, D, max_len);
}